// GCN_71451075936454
// MI455X (gfx1250) — compile-verified
//
#include <hip/hip_runtime.h>
#include <math.h>

// CDNA5 / gfx1250, wave32. FDIM = 256 output features for both layers.
typedef float v2f __attribute__((ext_vector_type(2)));
typedef float v8f __attribute__((ext_vector_type(8)));
typedef int   i32x4 __attribute__((vector_size(16)));

#define NFEAT 1024
#define FDIM  256
#define KT    32            // K-tile staged in LDS per iteration
#define EPS   1e-5f

// gfx1250 async global->LDS copy path (ASYNCcnt), guarded so either toolchain compiles.
#if defined(__has_builtin)
#if __has_builtin(__builtin_amdgcn_global_load_async_to_lds_b128) && \
    __has_builtin(__builtin_amdgcn_s_wait_asynccnt)
#define HAVE_ASYNC_LDS 1
#endif
#endif

#if defined(HAVE_ASYNC_LDS)
// Builtin expects (AS1 int4*, AS3 int4*, imm offset, imm cpol) per hipcc diagnostic.
typedef __attribute__((address_space(1))) i32x4* as1_v4i_ptr;
typedef __attribute__((address_space(3))) i32x4* as3_v4i_ptr;
#define WAIT_ASYNC(n) __builtin_amdgcn_s_wait_asynccnt(n)
#else
#define WAIT_ASYNC(n)
#endif

// Stage one contiguous KT*256-float tile of B (32 KB) into LDS.
// 256 threads * 8 * float4 == 8192 floats.
__device__ __forceinline__ void stage_tile(const float* __restrict__ gsrc,
                                           float* lds_dst, int tid) {
#if defined(HAVE_ASYNC_LDS)
#pragma unroll
  for (int i = 0; i < 8; ++i) {
    int idx = tid + i * 256;                       // float4 index
    __builtin_amdgcn_global_load_async_to_lds_b128(
        (as1_v4i_ptr)(gsrc + 4 * idx), (as3_v4i_ptr)(lds_dst + 4 * idx), 0, 0);
  }
#else
#pragma unroll
  for (int i = 0; i < 8; ++i) {
    int idx = tid + i * 256;
    *(float4*)(lds_dst + 4 * idx) = *(const float4*)(gsrc + 4 * idx);
  }
#endif
}

// ---------------------------------------------------------------- utilities
__global__ void fill_f32(float* __restrict__ p, size_t n, float v) {
  size_t i = (size_t)blockIdx.x * blockDim.x + threadIdx.x;
  size_t stride = (size_t)gridDim.x * blockDim.x;
  for (; i < n; i += stride) p[i] = v;
}

__global__ void deg_accum(const int* __restrict__ dst, const float* __restrict__ ew,
                          float* __restrict__ deg, int E) {
  int e = blockIdx.x * blockDim.x + threadIdx.x;
  if (e < E) atomicAdd(&deg[dst[e]], ew[e]);
}

__global__ void compute_dinv(const float* __restrict__ deg, float* __restrict__ dinv, int N) {
  int i = blockIdx.x * blockDim.x + threadIdx.x;
  if (i < N) {
    float d = deg[i] + 1.0f;                     // + self-loop weight 1
    dinv[i] = (d > 0.f) ? rsqrtf(d) : 0.f;
  }
}

// ------------------------------------------------- fp32 WMMA GEMM, N == 256
// C[M,256] = A[M,K] @ B[K,256].  Block = 256 threads = 8 waves, owns 32 rows.
// Waves 0-3 -> row-tile 0, waves 4-7 -> row-tile 1; wave's column strip = 64.
// B K-tiles (KT x 256, contiguous 32 KB) are double-buffered in LDS via
// async global->LDS b128 copies; each wave waits its own ASYNCcnt, then a
// workgroup barrier publishes all waves' tiles.
//
// WMMA fp32 16x16x4 VGPR layouts (CDNA5 ISA):
//  A: lane<16 -> M=lane, v0=K0,v1=K1 ; lane>=16 -> K2,K3    (float2 per lane)
//  B: lane%16 = N, v0 holds K = 2*(lane/16), v1 = K+1
//  C/D: VGPR i -> M = i + 8*(lane/16), N = lane%16
__global__ __launch_bounds__(256) void wmma_gemm_n256(
    const float* __restrict__ A, const float* __restrict__ B,
    float* __restrict__ C, int M, int K) {
  __shared__ __align__(16) float ldsB[2][KT * FDIM];   // 2 x 32 KB

  const int tid   = threadIdx.x;
  const int lane  = tid & 31;
  const int wave  = tid >> 5;
  const int half  = lane >> 4;
  const int lr    = lane & 15;
  const int rtile = wave >> 2;            // 0 or 1
  const int n0    = (wave & 3) * 64;
  const int row0  = blockIdx.x * 32;

  int arow = row0 + rtile * 16 + lr;      // clamp keeps EXEC uniform for WMMA
  if (arow >= M) arow = M - 1;
  const float* Arow = A + (size_t)arow * K;

  v8f acc0 = {}, acc1 = {}, acc2 = {}, acc3 = {};
  const int T = K / KT;

  stage_tile(B, ldsB[0], tid);            // prologue: tile 0 in flight

  for (int t = 0; t < T; ++t) {
    if (t + 1 < T) {                      // uniform branch
      stage_tile(B + (size_t)(t + 1) * KT * FDIM, ldsB[(t + 1) & 1], tid);
      WAIT_ASYNC(8);                      // tile t complete; t+1's 8 outstanding
    } else {
      WAIT_ASYNC(0);
    }
    __syncthreads();                      // publish tile t across waves

    const float* Bl = &ldsB[t & 1][0];
    const float* Ak = Arow + t * KT;
#pragma unroll
    for (int kk = 0; kk < KT; kk += 4) {
      v2f a = *(const v2f*)(Ak + kk + 2 * half);            // global_load_b64
      const float* Bp = Bl + (kk + 2 * half) * FDIM + n0 + lr;  // ds loads
      v2f b0 = {Bp[0],  Bp[FDIM]};
      v2f b1 = {Bp[16], Bp[FDIM + 16]};
      v2f b2 = {Bp[32], Bp[FDIM + 32]};
      v2f b3 = {Bp[48], Bp[FDIM + 48]};
      acc0 = __builtin_amdgcn_wmma_f32_16x16x4_f32(false, a, false, b0, (short)0, acc0, false, false);
      acc1 = __builtin_amdgcn_wmma_f32_16x16x4_f32(false, a, false, b1, (short)0, acc1, false, false);
      acc2 = __builtin_amdgcn_wmma_f32_16x16x4_f32(false, a, false, b2, (short)0, acc2, false, false);
      acc3 = __builtin_amdgcn_wmma_f32_16x16x4_f32(false, a, false, b3, (short)0, acc3, false, false);
    }
    __syncthreads();                      // tile t consumed; its buffer reusable
  }

#pragma unroll
  for (int i = 0; i < 8; ++i) {
    int r = row0 + rtile * 16 + i + 8 * half;
    if (r < M) {
      float* Cr = C + (size_t)r * FDIM + n0 + lr;
      Cr[0]  = acc0[i];
      Cr[16] = acc1[i];
      Cr[32] = acc2[i];
      Cr[48] = acc3[i];
    }
  }
}

// ------------------------------------------- edge scatter: out[dst] += n*h[src]
// One wave per edge (8 edges per 256-thread block). Gathers and atomics are
// lane-coalesced over the 256-float feature row; 205M global_atomic_add_f32
// total resolve in L2.
__global__ __launch_bounds__(256) void gcn_aggregate(
    const float* __restrict__ h, const int* __restrict__ src,
    const int* __restrict__ dst, const float* __restrict__ ew,
    const float* __restrict__ dinv, float* __restrict__ out, int E) {
  int e = blockIdx.x * 8 + (threadIdx.x >> 5);
  if (e >= E) return;
  int lane = threadIdx.x & 31;
  int s = src[e], d = dst[e];
  float norm = dinv[s] * ew[e] * dinv[d];
  const float* hs = h + (size_t)s * FDIM;
  float* od = out + (size_t)d * FDIM;
#pragma unroll
  for (int j = 0; j < 8; ++j) {
    int f = j * 32 + lane;
    atomicAdd(&od[f], norm * hs[f]);
  }
}

// out = relu(agg + dinv^2 * h (self-loop) + bias).  out may alias agg or h.
__global__ __launch_bounds__(256) void bias_selfloop_relu(
    const float* __restrict__ agg, const float* __restrict__ h,
    const float* __restrict__ dinv, const float* __restrict__ bias,
    float* __restrict__ out, int N) {
  size_t idx = (size_t)blockIdx.x * blockDim.x + threadIdx.x;
  if (idx < (size_t)N * FDIM) {
    int row = (int)(idx / FDIM);
    int f   = (int)(idx % FDIM);
    float di = dinv[row];
    float v = agg[idx] + di * di * h[idx] + bias[f];
    out[idx] = v > 0.f ? v : 0.f;
  }
}

// Per-feature sum / sumsq partials: thread t owns column t over a 256-row slab.
__global__ __launch_bounds__(256) void bn_stats(
    const float* __restrict__ h, float* __restrict__ sum,
    float* __restrict__ sumsq, int N) {
  int f = threadIdx.x;
  int r0 = blockIdx.x * 256;
  int r1 = min(r0 + 256, N);
  float s = 0.f, q = 0.f;
  for (int r = r0; r < r1; ++r) {
    float v = h[(size_t)r * FDIM + f];
    s += v; q += v * v;
  }
  atomicAdd(&sum[f], s);
  atomicAdd(&sumsq[f], q);
}

// Fused BatchNorm (batch stats) + LayerNorm over the 256 features of one row.
__global__ __launch_bounds__(256) void bn_ln_final(
    const float* __restrict__ h, const float* __restrict__ sum,
    const float* __restrict__ sumsq,
    const float* __restrict__ bn_g, const float* __restrict__ bn_b,
    const float* __restrict__ ln_g, const float* __restrict__ ln_b,
    float* __restrict__ out, int N) {
  __shared__ float s1[256], s2[256];
  int r = blockIdx.x, f = threadIdx.x;
  float invN = 1.0f / (float)N;
  float mu  = sum[f] * invN;
  float var = sumsq[f] * invN - mu * mu;
  var = var < 0.f ? 0.f : var;                         // biased variance
  float v = (h[(size_t)r * FDIM + f] - mu) * rsqrtf(var + EPS) * bn_g[f] + bn_b[f];
  s1[f] = v; s2[f] = v * v;
  __syncthreads();
  for (int off = 128; off > 0; off >>= 1) {
    if (f < off) { s1[f] += s1[f + off]; s2[f] += s2[f + off]; }
    __syncthreads();
  }
  float lmu  = s1[0] * (1.0f / FDIM);
  float lvar = s2[0] * (1.0f / FDIM) - lmu * lmu;
  lvar = lvar < 0.f ? 0.f : lvar;
  out[(size_t)r * FDIM + f] = (v - lmu) * rsqrtf(lvar + EPS) * ln_g[f] + ln_b[f];
}

// ---------------------------------------------------------------- launcher
extern "C" void kernel_launch(void* const* d_in, const int* in_sizes, int n_in,
                              void* d_out, int out_size, void* d_ws, size_t ws_size,
                              hipStream_t stream) {
  const float* x    = (const float*)d_in[0];
  const int*   ei   = (const int*)d_in[1];     // int32 (jax x64 disabled)
  const float* ew   = (const float*)d_in[2];
  const float* W1   = (const float*)d_in[3];
  const float* b1   = (const float*)d_in[4];
  const float* W2   = (const float*)d_in[5];
  const float* b2   = (const float*)d_in[6];
  const float* bn_g = (const float*)d_in[7];
  const float* bn_b = (const float*)d_in[8];
  const float* ln_g = (const float*)d_in[9];
  const float* ln_b = (const float*)d_in[10];

  const int N = in_sizes[0] / NFEAT;           // 50000
  const int E = in_sizes[2];                   // 800000
  const int* src = ei;
  const int* dst = ei + E;

  float* base = (float*)d_ws;
  size_t nf = (size_t)N * FDIM;
  float* bufA = base;                          // H1 / h1 / agg2 / h2
  float* bufB = base + nf;                     // agg1 / H2
  float* deg  = base + 2 * nf;
  float* dinv = deg + N;
  float* bns  = dinv + N;                      // 256 sums + 256 sumsqs
  float* out  = (float*)d_out;

  const int aggGrid = (E + 7) / 8;

  // Normalization coefficients
  fill_f32<<<256, 256, 0, stream>>>(deg, (size_t)N, 0.f);
  deg_accum<<<(E + 255) / 256, 256, 0, stream>>>(dst, ew, deg, E);
  compute_dinv<<<(N + 255) / 256, 256, 0, stream>>>(deg, dinv, N);

  // Layer 1: H = X @ W1 ; aggregate ; relu(agg + selfloop + b1)
  wmma_gemm_n256<<<(N + 31) / 32, 256, 0, stream>>>(x, W1, bufA, N, NFEAT);
  fill_f32<<<2048, 256, 0, stream>>>(bufB, nf, 0.f);
  gcn_aggregate<<<aggGrid, 256, 0, stream>>>(bufA, src, dst, ew, dinv, bufB, E);
  bias_selfloop_relu<<<N, 256, 0, stream>>>(bufB, bufA, dinv, b1, bufA, N);

  // Layer 2: H2 = h1 @ W2 ; aggregate ; relu(agg + selfloop + b2)
  wmma_gemm_n256<<<(N + 31) / 32, 256, 0, stream>>>(bufA, W2, bufB, N, FDIM);
  fill_f32<<<2048, 256, 0, stream>>>(bufA, nf, 0.f);
  gcn_aggregate<<<aggGrid, 256, 0, stream>>>(bufB, src, dst, ew, dinv, bufA, E);
  bias_selfloop_relu<<<N, 256, 0, stream>>>(bufA, bufB, dinv, b2, bufA, N);

  // BatchNorm stats + fused BN/LN epilogue
  fill_f32<<<1, 512, 0, stream>>>(bns, 512, 0.f);
  bn_stats<<<(N + 255) / 256, 256, 0, stream>>>(bufA, bns, bns + 256, N);
  bn_ln_final<<<N, 256, 0, stream>>>(bufA, bns, bns + 256,
                                     bn_g, bn_b, ln_g, ln_b, out, N);
}